// Synthesizing_75222057222524
// MI455X (gfx1250) — compile-verified
//
#include <hip/hip_runtime.h>
#include <stdint.h>

// out[g,c,h,w] = sum_s w[s,c] * X[g*4+s, c, h, w]
// Memory-bound streaming kernel (~566 MB traffic, ~24 us floor at 23.3 TB/s).
// CDNA5 path: async global->LDS b128 double-buffered pipeline + s_wait_asynccnt,
// b128 everywhere, non-temporal b128 stores. Each thread consumes only the LDS
// slots it loaded itself -> no workgroup barriers needed.

typedef float f32x4 __attribute__((ext_vector_type(4)));

#define NSAMP   4
#define NCH     3
#define HW4     18432          // (384*192)/4 float4 per (g,c) plane
#define TILE    256            // float4 per stage per sample (one per thread)
#define NITER   8              // stages per block
#define CHUNKS  9              // HW4 / (TILE*NITER) == 18432/2048

__device__ __forceinline__ void async_load_b128(const f32x4* gsrc, f32x4* ldst) {
#if defined(__HIP_DEVICE_COMPILE__) && defined(__gfx1250__)
  // VDST = LDS byte offset (low 32 bits of the generic shared address),
  // VADDR = 64-bit global address, GV mode (saddr = off). Tracked by ASYNCcnt.
  unsigned lds_off = (unsigned)(uintptr_t)ldst;
  unsigned long long gaddr = (unsigned long long)(uintptr_t)gsrc;
  asm volatile("global_load_async_to_lds_b128 %0, %1, off"
               :: "v"(lds_off), "v"(gaddr)
               : "memory");
#else
  *ldst = *gsrc;
#endif
}

__device__ __forceinline__ void wait_async_le4() {
#if defined(__HIP_DEVICE_COMPILE__) && defined(__gfx1250__)
  asm volatile("s_wait_asynccnt 0x4" ::: "memory");   // prev stage's 4 loads done
#endif
}

__device__ __forceinline__ void wait_async_0() {
#if defined(__HIP_DEVICE_COMPILE__) && defined(__gfx1250__)
  asm volatile("s_wait_asynccnt 0x0" ::: "memory");
#endif
}

__global__ __launch_bounds__(TILE) void wsum4_kernel(
    const float* __restrict__ Xf, const float* __restrict__ Wf,
    float* __restrict__ Of) {
  __shared__ f32x4 smem[2][NSAMP][TILE];   // 32 KB: double buffer x 4 samples

  const int tid   = threadIdx.x;
  const int bid   = blockIdx.x;
  const int chunk = bid % CHUNKS;
  const int plane = bid / CHUNKS;          // g*NCH + c
  const int c     = plane % NCH;
  const int g     = plane / NCH;

  // Block-uniform weights -> scalar loads / SGPR broadcast.
  const float w0 = Wf[0 * NCH + c];
  const float w1 = Wf[1 * NCH + c];
  const float w2 = Wf[2 * NCH + c];
  const float w3 = Wf[3 * NCH + c];

  const f32x4* in4  = (const f32x4*)Xf;
  f32x4*       out4 = (f32x4*)Of;

  const size_t chunkOff = (size_t)chunk * (TILE * NITER);
  const f32x4* src0 = in4 + ((size_t)((g * NSAMP + 0) * NCH + c) * HW4 + chunkOff);
  const f32x4* src1 = in4 + ((size_t)((g * NSAMP + 1) * NCH + c) * HW4 + chunkOff);
  const f32x4* src2 = in4 + ((size_t)((g * NSAMP + 2) * NCH + c) * HW4 + chunkOff);
  const f32x4* src3 = in4 + ((size_t)((g * NSAMP + 3) * NCH + c) * HW4 + chunkOff);
  f32x4* dst = out4 + ((size_t)plane * HW4 + chunkOff);

  // Prologue: stage 0 into buffer 0.
  async_load_b128(src0 + tid, &smem[0][0][tid]);
  async_load_b128(src1 + tid, &smem[0][1][tid]);
  async_load_b128(src2 + tid, &smem[0][2][tid]);
  async_load_b128(src3 + tid, &smem[0][3][tid]);

  for (int i = 0; i < NITER; ++i) {
    const int cur = i & 1;
    if (i + 1 < NITER) {
      const int off = (i + 1) * TILE + tid;
      async_load_b128(src0 + off, &smem[cur ^ 1][0][tid]);
      async_load_b128(src1 + off, &smem[cur ^ 1][1][tid]);
      async_load_b128(src2 + off, &smem[cur ^ 1][2][tid]);
      async_load_b128(src3 + off, &smem[cur ^ 1][3][tid]);
      wait_async_le4();      // ASYNCcnt <= 4: stage i data has landed in LDS
    } else {
      wait_async_0();
    }

    f32x4 a = smem[cur][0][tid];
    f32x4 b = smem[cur][1][tid];
    f32x4 e = smem[cur][2][tid];
    f32x4 d = smem[cur][3][tid];
    f32x4 r = a * w0 + b * w1 + e * w2 + d * w3;

    // Output is written once, never re-read: bypass-friendly NT store.
    __builtin_nontemporal_store(r, dst + i * TILE + tid);
  }
}

extern "C" void kernel_launch(void* const* d_in, const int* in_sizes, int n_in,
                              void* d_out, int out_size, void* d_ws, size_t ws_size,
                              hipStream_t stream) {
  (void)n_in; (void)out_size; (void)d_ws; (void)ws_size;
  const float* X = (const float*)d_in[0];
  const float* W = (const float*)d_in[1];
  float*       O = (float*)d_out;

  // groups = B / NSAMP, derived from the flat input size (B*3*384*192 floats)
  const int groups = in_sizes[0] / (NSAMP * NCH * HW4 * 4);
  const int grid   = groups * NCH * CHUNKS;   // 128*3*9 = 3456 blocks

  wsum4_kernel<<<grid, TILE, 0, stream>>>(X, W, O);
}